// ExpandNodesLayer_84473416778477
// MI455X (gfx1250) — compile-verified
//
#include <hip/hip_runtime.h>
#include <hip/hip_bf16.h>

// Problem constants (from the reference): N=64, E=64, H=768, L=512.
#define N_  64
#define E_  64
#define H_  768
#define L_  512
#define LT_ (L_ / 16)   // 32 L-tiles per batch
#define TPW 6           // 16-wide H-tiles per wave: 8 waves * 6 * 16 = 768 = H
#define EPS_ 1e-12f

typedef __attribute__((ext_vector_type(2))) float v2f;
typedef __attribute__((ext_vector_type(8))) float v8f;

// Fused  expand[n,l,h] = sum_e mapping[n,e,l] * state[n,e,h]  +  LayerNorm over h.
// One workgroup (8 wave32) = one (n, 16-row L-tile); each wave covers 96 columns of H.
__global__ __launch_bounds__(256)
void ExpandNodesLN_kernel(const float* __restrict__ state,    // (N,E,H)
                          const float* __restrict__ mapping,  // (N,E,L)
                          const float* __restrict__ gamma,    // (H)
                          const float* __restrict__ beta,     // (H)
                          float* __restrict__ out) {          // (N,L,H)
  __shared__ float s_sum[16];
  __shared__ float s_sq[16];

  const int n    = blockIdx.x / LT_;
  const int l0   = (blockIdx.x % LT_) * 16;
  const int lane = threadIdx.x & 31;
  const int wave = threadIdx.x >> 5;
  const int ln16 = lane & 15;
  const int half = lane >> 4;              // which half-wave (K split / row split)

  if (threadIdx.x < 16) { s_sum[threadIdx.x] = 0.0f; s_sq[threadIdx.x] = 0.0f; }

  // A-matrix source: A[l, e] = mapping[n, e, l0 + (lane&15)], stride L between e's.
  const float* Mb = mapping + (size_t)n * E_ * L_ + l0 + ln16;
  // B-matrix source: B[e, h] = state[n, e, h], h contiguous.
  const float* Sb = state + (size_t)n * E_ * H_;
  const int hbase = wave * (TPW * 16);

  v8f acc[TPW] = {};

  // K loop: V_WMMA_F32_16X16X4_F32, 16 steps of K=4 cover E=64.
  // ISA 32-bit A layout: VGPR0 = {K=k | K=k+2}, VGPR1 = {K=k+1 | K=k+3} per half-wave;
  // B mirrors the same K split across half-waves with N = lane&15.
  for (int k = 0; k < E_; k += 4) {
    const int e = k + (half << 1);
    v2f a;
    a.x = Mb[(size_t)e * L_];
    a.y = Mb[(size_t)(e + 1) * L_];
#pragma unroll
    for (int t = 0; t < TPW; ++t) {
      const int h = hbase + t * 16 + ln16;
      v2f b;
      b.x = Sb[(size_t)e * H_ + h];
      b.y = Sb[(size_t)(e + 1) * H_ + h];
      // 8 args: (neg_a, A, neg_b, B, c_mod, C, reuse_a, reuse_b)
      acc[t] = __builtin_amdgcn_wmma_f32_16x16x4_f32(
          false, a, false, b, (short)0, acc[t], false, false);
    }
  }

  // ---- LayerNorm reduction over H -------------------------------------------
  // C layout: VGPR m of a tile holds row (m + 8*half), column (lane&15)+tile*16.
  float ps[8], pq[8];
#pragma unroll
  for (int m = 0; m < 8; ++m) {
    float s = 0.0f, q = 0.0f;
#pragma unroll
    for (int t = 0; t < TPW; ++t) {
      const float v = acc[t][m];
      s += v;
      q += v * v;
    }
    // Reduce across the 16 lanes of this half-wave (masks < 16 keep halves apart).
#pragma unroll
    for (int mask = 1; mask < 16; mask <<= 1) {
      s += __shfl_xor(s, mask, 32);
      q += __shfl_xor(q, mask, 32);
    }
    ps[m] = s;
    pq[m] = q;
  }

  __syncthreads();  // s_sum/s_sq init happens-before the atomics
  if (ln16 == 0) {
#pragma unroll
    for (int m = 0; m < 8; ++m) {
      const int row = m + (half << 3);
      atomicAdd(&s_sum[row], ps[m]);   // ds_add_f32
      atomicAdd(&s_sq[row],  pq[m]);
    }
  }
  __syncthreads();

  const float invH = 1.0f / (float)H_;
  float mu_[8], ri_[8];
#pragma unroll
  for (int m = 0; m < 8; ++m) {
    const int row = m + (half << 3);
    const float mu  = s_sum[row] * invH;
    const float var = s_sq[row] * invH - mu * mu;
    mu_[m] = mu;
    ri_[m] = rsqrtf(var + EPS_);
  }

  // ---- Scale/shift + store ---------------------------------------------------
  float* Ob = out + (size_t)n * L_ * H_ + (size_t)l0 * H_;
#pragma unroll
  for (int t = 0; t < TPW; ++t) {
    const int h = hbase + t * 16 + ln16;
    const float g = gamma[h];
    const float bb = beta[h];
#pragma unroll
    for (int m = 0; m < 8; ++m) {
      const int row = m + (half << 3);
      Ob[(size_t)row * H_ + h] = (acc[t][m] - mu_[m]) * ri_[m] * g + bb;
    }
  }
}

extern "C" void kernel_launch(void* const* d_in, const int* in_sizes, int n_in,
                              void* d_out, int out_size, void* d_ws, size_t ws_size,
                              hipStream_t stream) {
  (void)in_sizes; (void)n_in; (void)out_size; (void)d_ws; (void)ws_size;
  const float* state   = (const float*)d_in[0];  // (N,E,H)
  const float* mapping = (const float*)d_in[1];  // (N,E,L)
  const float* gamma   = (const float*)d_in[2];  // (H)
  const float* beta    = (const float*)d_in[3];  // (H)
  float* out = (float*)d_out;                    // (N,L,H)

  dim3 grid(N_ * LT_);
  dim3 block(256);
  ExpandNodesLN_kernel<<<grid, block, 0, stream>>>(state, mapping, gamma, beta, out);
}